// MMD_loss_56667798504269
// MI455X (gfx1250) — compile-verified
//
#include <hip/hip_runtime.h>
#include <math.h>

typedef float v2f __attribute__((ext_vector_type(2)));
typedef float v8f __attribute__((ext_vector_type(8)));

#define N_HALF 4096
#define NS     8192
#define KDIM   256
#define TILE   128
#define KC     16
#define LSTR   20   // KC + 4 pad: 20*l mod 64 -> distinct multiples of 4 -> conflict-free b64 frag loads
#define BUFSZ  (TILE * LSTR)
#define NCH    (KDIM / KC)

__device__ __forceinline__ const float* trow(const float* __restrict__ s,
                                             const float* __restrict__ t, int i) {
  return (i < N_HALF) ? (s + (size_t)i * KDIM) : (t + (size_t)(i - N_HALF) * KDIM);
}

// CDNA5 async global->LDS DMA (no VGPR round-trip; tracked by ASYNCcnt)
__device__ __forceinline__ void async_ld_b128(unsigned lds_byte_off, const float* g) {
  asm volatile("global_load_async_to_lds_b128 %0, %1, off"
               :: "v"(lds_byte_off), "v"(g) : "memory");
}
__device__ __forceinline__ void wait_async0() {
  asm volatile("s_wait_asynccnt 0x0" ::: "memory");
}

// ---------- pass 0: zero the colsum accumulator ----------
__global__ void zero_kernel(float* __restrict__ s) {
  s[threadIdx.x] = 0.0f;
}

// ---------- pass 1a: per-row sum of squares ----------
__global__ void rowsq_kernel(const float* __restrict__ src, const float* __restrict__ tgt,
                             float* __restrict__ sq) {
  int i = blockIdx.x * 256 + threadIdx.x;
  const float4* p = (const float4*)trow(src, tgt, i);
  float acc = 0.0f;
  #pragma unroll 8
  for (int c = 0; c < KDIM / 4; ++c) {
    float4 v = p[c];
    acc += v.x * v.x + v.y * v.y + v.z * v.z + v.w * v.w;
  }
  sq[i] = acc;
}

// ---------- pass 1b: column sums (for sum(G) = ||colsum||^2) ----------
__global__ void colsum_kernel(const float* __restrict__ src, const float* __restrict__ tgt,
                              float* __restrict__ s) {
  int row0 = blockIdx.x * 128;
  int k = threadIdx.x;
  float acc = 0.0f;
  for (int r = 0; r < 128; ++r) acc += trow(src, tgt, row0 + r)[k];
  atomicAdd(&s[k], acc);
}

// ---------- pass 2: bandwidth -> 5 negated inverse denominators ----------
__global__ void bw_kernel(const float* __restrict__ sq, const float* __restrict__ s,
                          float* __restrict__ ni) {
  int t = threadIdx.x, lane = t & 31, w = t >> 5;
  float a = 0.0f;
  #pragma unroll
  for (int j = 0; j < NS / 256; ++j) a += sq[t + 256 * j];
  float b = s[t] * s[t];
  #pragma unroll
  for (int m = 16; m; m >>= 1) {
    a += __shfl_xor(a, m, 32);
    b += __shfl_xor(b, m, 32);
  }
  __shared__ float ra[8], rb[8];
  if (lane == 0) { ra[w] = a; rb[w] = b; }
  __syncthreads();
  if (t == 0) {
    double sumsq = 0.0, s2 = 0.0;
    for (int i = 0; i < 8; ++i) { sumsq += (double)ra[i]; s2 += (double)rb[i]; }
    // sum(dist) = 2*ns*sum(sq) - 2*||colsum||^2 ; bandwidth = sum(dist)/(ns^2-ns)/2^(5//2)
    double sumdist = 2.0 * (double)NS * sumsq - 2.0 * s2;
    double bw = sumdist / ((double)NS * (double)NS - (double)NS) / 4.0;
    for (int q = 0; q < 5; ++q) ni[q] = (float)(-1.0 / (bw * (double)(1 << q)));
  }
}

// ---------- pass 3: fused WMMA GEMM tile + 5-kernel exp + signed reduce ----------
__global__ __launch_bounds__(256) void mmd_tile_kernel(
    const float* __restrict__ src, const float* __restrict__ tgt,
    const float* __restrict__ sq, const float* __restrict__ ni,
    float* __restrict__ partials) {
  __shared__ float lA[2][BUFSZ];
  __shared__ float lB[2][BUFSZ];
  __shared__ float red[8];

  const int bi = blockIdx.x, bj = blockIdx.y;
  const int tid = threadIdx.x;
  const int w = tid >> 5, lane = tid & 31, l15 = lane & 15, hf = lane >> 4;

  const unsigned offA = (unsigned)(uintptr_t)&lA[0][0];  // low 32 bits = LDS byte offset
  const unsigned offB = (unsigned)(uintptr_t)&lB[0][0];

  v8f c[8];
  v8f zero = 0.0f;
  #pragma unroll
  for (int t = 0; t < 8; ++t) c[t] = zero;

  const int arow0 = bi * TILE, brow0 = bj * TILE;

  // per-thread staging slots: 2 float4 per tile (128x16 fp32 tile / 256 threads)
  const int e0 = tid, e1 = tid + 256;
  const int row0s = e0 >> 2, c40 = (e0 & 3) << 2;
  const int row1s = e1 >> 2, c41 = (e1 & 3) << 2;
  const float* ga0 = trow(src, tgt, arow0 + row0s) + c40;
  const float* ga1 = trow(src, tgt, arow0 + row1s) + c41;
  const float* gb0 = trow(src, tgt, brow0 + row0s) + c40;
  const float* gb1 = trow(src, tgt, brow0 + row1s) + c41;
  const unsigned la0 = (unsigned)((row0s * LSTR + c40) * 4);
  const unsigned la1 = (unsigned)((row1s * LSTR + c41) * 4);

  // prologue: chunk 0 -> buffer 0
  async_ld_b128(offA + la0, ga0);
  async_ld_b128(offA + la1, ga1);
  async_ld_b128(offB + la0, gb0);
  async_ld_b128(offB + la1, gb1);

  for (int ch = 0; ch < NCH; ++ch) {
    wait_async0();          // own async writes for current chunk landed in LDS
    __syncthreads();        // everyone's landed; previous buffer fully consumed
    const int buf = ch & 1;
    if (ch + 1 < NCH) {     // prefetch next chunk into the other buffer (overlaps WMMAs)
      const int k1 = (ch + 1) * KC;
      const unsigned bb = (unsigned)((buf ^ 1) * BUFSZ * 4);
      async_ld_b128(offA + bb + la0, ga0 + k1);
      async_ld_b128(offA + bb + la1, ga1 + k1);
      async_ld_b128(offB + bb + la0, gb0 + k1);
      async_ld_b128(offB + bb + la1, gb1 + k1);
    }
    #pragma unroll
    for (int ks = 0; ks < KC / 4; ++ks) {
      const int koff = (ks << 2) + (hf << 1); // lanes 0-15: K={k,k+1}, lanes 16-31: K={k+2,k+3}
      v2f a = *(const v2f*)&lA[buf][(w * 16 + l15) * LSTR + koff];
      #pragma unroll
      for (int t = 0; t < 8; ++t) {
        v2f b = *(const v2f*)&lB[buf][(t * 16 + l15) * LSTR + koff];
        // D = A(16x4 f32) x B(4x16 f32) + C : exact fp32 Gram accumulation
        c[t] = __builtin_amdgcn_wmma_f32_16x16x4_f32(
            false, a, false, b, (short)0, c[t], false, false);
      }
    }
  }

  // epilogue: dist -> 5 exponentials -> signed accumulate
  float nib[5];
  #pragma unroll
  for (int q = 0; q < 5; ++q) nib[q] = ni[q];

  float sqi[8], sqj[8];
  const int ib = arow0 + w * 16 + 8 * hf;   // C/D layout: VGPR v -> row v (lanes 0-15) / v+8 (16-31)
  #pragma unroll
  for (int v = 0; v < 8; ++v) sqi[v] = sq[ib + v];
  const int jb = brow0 + l15;               // col = lane & 15 within each 16-wide tile
  #pragma unroll
  for (int t = 0; t < 8; ++t) sqj[t] = sq[jb + t * 16];

  float acc = 0.0f;
  #pragma unroll
  for (int t = 0; t < 8; ++t) {
    #pragma unroll
    for (int v = 0; v < 8; ++v) {
      float d = sqi[v] + sqj[t] - 2.0f * c[t][v];
      float e = 0.0f;
      #pragma unroll
      for (int q = 0; q < 5; ++q) e += __expf(d * nib[q]);
      acc += e;
    }
  }
  #pragma unroll
  for (int m = 16; m; m >>= 1) acc += __shfl_xor(acc, m, 32);
  if (lane == 0) red[w] = acc;
  __syncthreads();
  if (tid == 0) {
    float sum = 0.0f;
    #pragma unroll
    for (int i = 0; i < 8; ++i) sum += red[i];
    // n = 4096 is tile-aligned -> quadrant sign is uniform per block
    float sgn = ((bi < 32) == (bj < 32)) ? 1.0f : -1.0f;
    partials[bi * gridDim.y + bj] = sgn * sum;
  }
}

// ---------- pass 4: final reduce of 4096 block partials ----------
__global__ void finish_kernel(const float* __restrict__ partials, float* __restrict__ out) {
  int t = threadIdx.x, lane = t & 31, w = t >> 5;
  double a = 0.0;
  #pragma unroll
  for (int j = 0; j < 4096 / 256; ++j) a += (double)partials[t + 256 * j];
  #pragma unroll
  for (int m = 16; m; m >>= 1) a += __shfl_xor(a, m, 32);
  __shared__ double r[8];
  if (lane == 0) r[w] = a;
  __syncthreads();
  if (t == 0) {
    double s = 0.0;
    for (int i = 0; i < 8; ++i) s += r[i];
    out[0] = (float)(s / 5.0 / ((double)N_HALF * (double)N_HALF) / 2.0);
  }
}

extern "C" void kernel_launch(void* const* d_in, const int* in_sizes, int n_in,
                              void* d_out, int out_size, void* d_ws, size_t ws_size,
                              hipStream_t stream) {
  const float* src = (const float*)d_in[0];
  const float* tgt = (const float*)d_in[1];
  float* ws = (float*)d_ws;
  float* sq = ws;          // 8192 floats: row sum-of-squares
  float* s  = ws + 8192;   // 256 floats:  column sums
  float* ni = ws + 8448;   // 8 floats:    5 negated inverse bandwidths
  float* pt = ws + 8456;   // 4096 floats: per-block signed partials

  zero_kernel  <<<1,        256, 0, stream>>>(s);
  rowsq_kernel <<<NS / 256, 256, 0, stream>>>(src, tgt, sq);
  colsum_kernel<<<NS / 128, 256, 0, stream>>>(src, tgt, s);
  bw_kernel    <<<1,        256, 0, stream>>>(sq, s, ni);
  dim3 grid(NS / TILE, NS / TILE);
  mmd_tile_kernel<<<grid,   256, 0, stream>>>(src, tgt, sq, ni, pt);
  finish_kernel<<<1,        256, 0, stream>>>(pt, (float*)d_out);
}